// fusion_9990093931251
// MI455X (gfx1250) — compile-verified
//
#include <hip/hip_runtime.h>

typedef __attribute__((ext_vector_type(2))) float v2f;
typedef __attribute__((ext_vector_type(8))) float v8f;

#define B_ 8
#define N_ 4096
#define C_ 1024
#define S_ 16                 // splits of the n-reduction in the WMMA kernel
#define NCHUNK_ (N_ / S_)     // 256 rows of n per wave

// ---------------------------------------------------------------------------
// Kernel 1: s2[b,n] = sum_d f2[b,n,d].  One wave per row, float4 loads.
// ---------------------------------------------------------------------------
__global__ void rowsum_f2(const float* __restrict__ f2, float* __restrict__ s2) {
    const int row  = blockIdx.x * (blockDim.x >> 5) + (threadIdx.x >> 5);
    const int lane = threadIdx.x & 31;
    if (row >= B_ * N_) return;
    const float4* p = (const float4*)(f2 + (size_t)row * C_);
    float acc = 0.f;
    #pragma unroll
    for (int i = lane; i < C_ / 4; i += 32) {
        float4 v = p[i];
        acc += (v.x + v.y) + (v.z + v.w);
    }
    #pragma unroll
    for (int off = 16; off >= 1; off >>= 1)
        acc += __shfl_xor(acc, off, 32);
    if (lane == 0) s2[row] = acc;
}

// ---------------------------------------------------------------------------
// Kernel 2: partial C_row via V_WMMA_F32_16X16X4_F32.
//   D = A(16x4) x B(4x16) + C, chained over the wave's n-chunk.
//   A[m,k] = f1[b, n0+k, c0+m]   (M = channel tile)
//   B[k,j] = s2[b, n0+k]         (all 16 columns broadcast s2)
//   => every column of D equals the partial C_row for this c-tile.
// A-layout (ISA 7.12.2, 32-bit A 16x4): lanes 0-15 hold K=0 (v0), K=1 (v1);
// lanes 16-31 hold K=2 (v0), K=3 (v1). B mirrors this K split.
// D column N=0: lane 0 = M=0..7 in acc[0..7]; lane 16 = M=8..15.
// ---------------------------------------------------------------------------
__global__ void matvec_wmma(const float* __restrict__ f1,
                            const float* __restrict__ s2,
                            float* __restrict__ partials) {
    const int c0   = blockIdx.x * 16;
    const int b    = blockIdx.y;
    const int s    = blockIdx.z;
    const int lane = threadIdx.x;       // 0..31, one wave per block
    const int half = lane >> 4;         // 0: K base 0, 1: K base 2
    const int cl   = lane & 15;         // channel within tile (M)

    const float* f1b = f1 + (size_t)b * N_ * C_;
    const float* s2b = s2 + b * N_;

    v8f acc = {};
    const int nbeg = s * NCHUNK_;
    #pragma unroll 4
    for (int n0 = nbeg; n0 < nbeg + NCHUNK_; n0 += 4) {
        const int nb = n0 + half * 2;
        v2f a, bv;
        a.x  = f1b[(size_t)nb       * C_ + c0 + cl];
        a.y  = f1b[(size_t)(nb + 1) * C_ + c0 + cl];
        bv.x = s2b[nb];
        bv.y = s2b[nb + 1];
        acc = __builtin_amdgcn_wmma_f32_16x16x4_f32(
            /*neg_a=*/false, a, /*neg_b=*/false, bv,
            /*c_mod=*/(short)0, acc, /*reuse_a=*/false, /*reuse_b=*/false);
    }

    // Extract column N=0 of D and store partial C_row for this (s, b, c-tile).
    if (cl == 0) {
        float* dst = partials + ((size_t)s * B_ + b) * C_ + c0 + half * 8;
        #pragma unroll
        for (int i = 0; i < 8; ++i) dst[i] = acc[i];
    }
}

// ---------------------------------------------------------------------------
// Kernel 3: alpha[b,c] = sigmoid( sum_s partials[s,b,c] ).  Deterministic.
// ---------------------------------------------------------------------------
__global__ void reduce_sigmoid(const float* __restrict__ partials,
                               float* __restrict__ alpha) {
    const int idx = blockIdx.x * blockDim.x + threadIdx.x;  // b*C + c
    if (idx >= B_ * C_) return;
    const int b = idx / C_;
    const int c = idx - b * C_;
    float sum = 0.f;
    #pragma unroll
    for (int s = 0; s < S_; ++s)
        sum += partials[((size_t)s * B_ + b) * C_ + c];
    alpha[idx] = 1.f / (1.f + __expf(-sum));
}

// ---------------------------------------------------------------------------
// Kernel 4: out[b,n,c] = f1[b,n,c] + alpha[b,c], float4-vectorized stream.
// ---------------------------------------------------------------------------
__global__ void add_alpha(const float* __restrict__ f1,
                          const float* __restrict__ alpha,
                          float* __restrict__ out) {
    const size_t i4 = (size_t)blockIdx.x * blockDim.x + threadIdx.x;
    const size_t total4 = (size_t)B_ * N_ * C_ / 4;
    if (i4 >= total4) return;
    const int c4 = (int)(i4 % (C_ / 4));
    const int b  = (int)(i4 / ((size_t)N_ * C_ / 4));
    const float4 a4 = *(const float4*)(alpha + (size_t)b * C_ + c4 * 4);
    float4 v = ((const float4*)f1)[i4];
    v.x += a4.x; v.y += a4.y; v.z += a4.z; v.w += a4.w;
    ((float4*)out)[i4] = v;
}

// ---------------------------------------------------------------------------
extern "C" void kernel_launch(void* const* d_in, const int* in_sizes, int n_in,
                              void* d_out, int out_size, void* d_ws, size_t ws_size,
                              hipStream_t stream) {
    const float* f1 = (const float*)d_in[0];
    const float* f2 = (const float*)d_in[1];
    float* out = (float*)d_out;

    // Workspace layout (floats): s2[B*N] | partials[S*B*C] | alpha[B*C]
    float* s2       = (float*)d_ws;
    float* partials = s2 + (size_t)B_ * N_;
    float* alpha    = partials + (size_t)S_ * B_ * C_;

    // 1) row sums of f2
    rowsum_f2<<<(B_ * N_) / 8, 256, 0, stream>>>(f2, s2);

    // 2) WMMA partial matvec: grid (c-tiles, batches, n-splits), 1 wave/block
    dim3 g2(C_ / 16, B_, S_);
    matvec_wmma<<<g2, 32, 0, stream>>>(f1, s2, partials);

    // 3) reduce partials + sigmoid
    reduce_sigmoid<<<(B_ * C_ + 255) / 256, 256, 0, stream>>>(partials, alpha);

    // 4) broadcast add
    const size_t total4 = (size_t)B_ * N_ * C_ / 4;
    add_alpha<<<(unsigned)((total4 + 255) / 256), 256, 0, stream>>>(f1, alpha, out);
}